// GNN_65498251264428
// MI455X (gfx1250) — compile-verified
//
#include <hip/hip_runtime.h>

#define NN 100000
#define NE 1000000
#define FDIM 64
#define NG 128

typedef float v2f __attribute__((ext_vector_type(2)));
typedef float v8f __attribute__((ext_vector_type(8)));

__global__ void zero_f32(float* __restrict__ p, long n) {
    long i = (long)blockIdx.x * blockDim.x + threadIdx.x;
    if (i < n) p[i] = 0.0f;
}

// One thread per (edge, feature). Consecutive 64 threads share an edge, so a
// wave32 issues a fully-coalesced 128B gather + 128B contiguous atomic burst.
__global__ void edge_scatter(const float* __restrict__ x,
                             const int* __restrict__ src,
                             const int* __restrict__ dst,
                             const float* __restrict__ ew,
                             float* __restrict__ aggr) {
    long idx = (long)blockIdx.x * blockDim.x + threadIdx.x;
    if (idx >= (long)NE * FDIM) return;
    int e = (int)(idx >> 6);
    int f = (int)(idx & 63);
    float v = x[(long)src[e] * FDIM + f] * ew[e];
    atomicAdd(&aggr[(long)dst[e] * FDIM + f], v);
}

// Fused: out = act(A1 @ Wrel^T + bias + A2 @ Wroot^T)
// Block = 128 threads = 4 waves; wave w computes output columns [16w, 16w+16)
// for the block's 16 rows. Full-precision V_WMMA_F32_16X16X4_F32, K=64 in 16 steps.
__global__ __launch_bounds__(128) void gemm_fused(
    const float* __restrict__ A1, const float* __restrict__ A2,
    const float* __restrict__ Wrel, const float* __restrict__ Wroot,
    const float* __restrict__ bias, float* __restrict__ out, int do_relu)
{
    const int wave = threadIdx.x >> 5;
    const int lane = threadIdx.x & 31;
    const int half = lane >> 4;     // K-pair selector for A/B operands
    const int l15  = lane & 15;     // row index (A) / column index (B, C/D)
    const int row0 = blockIdx.x * 16;
    const int col0 = wave * 16;

    // C/D layout: VGPR r, lane-half h -> row r+8h, col l15. Bias is per-column,
    // so every accumulator element of a lane starts at bias[col0+l15].
    v8f acc;
    const float bv = bias[col0 + l15];
#pragma unroll
    for (int i = 0; i < 8; ++i) acc[i] = bv;

    // A 16x4 f32 layout: lane half h supplies K = {2h, 2h+1} of row l15.
    // B 4x16: lane l15 = column; half h supplies K = {2h, 2h+1}. W is [hout][fin]
    // row-major, so the K-pair is a contiguous float2 in the weight row.
    const float* a1 = A1   + (long)(row0 + l15) * FDIM + 2 * half;
    const float* a2 = A2   + (long)(row0 + l15) * FDIM + 2 * half;
    const float* b1 = Wrel + (long)(col0 + l15) * FDIM + 2 * half;
    const float* b2 = Wroot + (long)(col0 + l15) * FDIM + 2 * half;

#pragma unroll
    for (int k = 0; k < FDIM; k += 4) {
        v2f a = *(const v2f*)(a1 + k);
        v2f b = *(const v2f*)(b1 + k);
        acc = __builtin_amdgcn_wmma_f32_16x16x4_f32(
            false, a, false, b, (short)0, acc, false, false);
    }
#pragma unroll
    for (int k = 0; k < FDIM; k += 4) {
        v2f a = *(const v2f*)(a2 + k);
        v2f b = *(const v2f*)(b2 + k);
        acc = __builtin_amdgcn_wmma_f32_16x16x4_f32(
            false, a, false, b, (short)0, acc, false, false);
    }

    if (do_relu) {
#pragma unroll
        for (int i = 0; i < 8; ++i) acc[i] = fmaxf(acc[i], 0.0f);
    }

#pragma unroll
    for (int r = 0; r < 8; ++r)
        out[(long)(row0 + r + 8 * half) * FDIM + col0 + l15] = acc[r];
}

__global__ void pool_scatter(const float* __restrict__ h,
                             const int* __restrict__ batch,
                             float* __restrict__ pooled,
                             float* __restrict__ counts) {
    long idx = (long)blockIdx.x * blockDim.x + threadIdx.x;
    if (idx >= (long)NN * FDIM) return;
    int n = (int)(idx >> 6);
    int f = (int)(idx & 63);
    int g = batch[n];
    atomicAdd(&pooled[g * FDIM + f], h[idx]);
    if (f == 0) atomicAdd(&counts[g], 1.0f);
}

__global__ void pool_finish(const float* __restrict__ pooled,
                            const float* __restrict__ counts,
                            const float* __restrict__ wlin,
                            const float* __restrict__ blin,
                            float* __restrict__ out) {
    int g = threadIdx.x;
    if (g >= NG) return;
    float c = fmaxf(counts[g], 1.0f);
    float s = 0.0f;
    for (int f = 0; f < FDIM; ++f) s += pooled[g * FDIM + f] * wlin[f];
    out[g] = s / c + blin[0];
}

extern "C" void kernel_launch(void* const* d_in, const int* in_sizes, int n_in,
                              void* d_out, int out_size, void* d_ws, size_t ws_size,
                              hipStream_t stream) {
    (void)in_sizes; (void)n_in; (void)out_size; (void)ws_size;

    const float* x       = (const float*)d_in[0];
    const int*   ei      = (const int*)d_in[1];
    const float* ew      = (const float*)d_in[2];
    const int*   batch   = (const int*)d_in[3];
    const float* w1_rel  = (const float*)d_in[4];
    const float* b1_rel  = (const float*)d_in[5];
    const float* w1_root = (const float*)d_in[6];
    const float* w2_rel  = (const float*)d_in[7];
    const float* b2_rel  = (const float*)d_in[8];
    const float* w2_root = (const float*)d_in[9];
    const float* w_lin   = (const float*)d_in[10];
    const float* b_lin   = (const float*)d_in[11];
    float* out = (float*)d_out;

    const int* src = ei;        // edge_index[0]
    const int* dst = ei + NE;   // edge_index[1]

    // Workspace: aggr (also h2; safe alias — each gemm block reads only its own
    // 16 rows before storing them), h1, pooled sums, counts.
    float* aggr   = (float*)d_ws;                 // NN*FDIM
    float* h1     = aggr + (long)NN * FDIM;       // NN*FDIM
    float* pooled = h1 + (long)NN * FDIM;         // NG*FDIM
    float* counts = pooled + NG * FDIM;           // NG
    float* h2     = aggr;

    const long nvals = (long)NN * FDIM;
    const long ethreads = (long)NE * FDIM;
    const int zgrid = (int)((nvals + 255) / 256);
    const int egrid = (int)((ethreads + 255) / 256);

    // ---- Layer 1 ----
    zero_f32<<<zgrid, 256, 0, stream>>>(aggr, nvals);
    edge_scatter<<<egrid, 256, 0, stream>>>(x, src, dst, ew, aggr);
    gemm_fused<<<NN / 16, 128, 0, stream>>>(aggr, x, w1_rel, w1_root, b1_rel, h1, 1);

    // ---- Layer 2 ----
    zero_f32<<<zgrid, 256, 0, stream>>>(aggr, nvals);
    edge_scatter<<<egrid, 256, 0, stream>>>(h1, src, dst, ew, aggr);
    gemm_fused<<<NN / 16, 128, 0, stream>>>(aggr, h1, w2_rel, w2_root, b2_rel, h2, 0);

    // ---- Pool + head ----
    zero_f32<<<(NG * FDIM + NG + 255) / 256, 256, 0, stream>>>(pooled, NG * FDIM + NG);
    pool_scatter<<<zgrid, 256, 0, stream>>>(h2, batch, pooled, counts);
    pool_finish<<<1, NG, 0, stream>>>(pooled, counts, w_lin, b_lin, out);
}